// SCAConv_DS_6425271074980
// MI455X (gfx1250) — compile-verified
//
#include <hip/hip_runtime.h>

// SCAConv fused kernel for gfx1250 (MI455X), wave32, fp32 WMMA 16x16x4.
// B=8, C_IN=C_OUT=32, H=W=128, K=3, C_LEN=8, HID=16.

#define B_   8
#define CI   32
#define CO   32
#define HH   128
#define WW   128
#define HID_ 16
#define TL   16      // locations per workgroup

typedef __attribute__((ext_vector_type(2))) float v2f;
typedef __attribute__((ext_vector_type(8))) float v8f;

__global__ __launch_bounds__(256)
void scaconv_ds_kernel(const float* __restrict__ x,        // (8,32,128,128)
                       const float* __restrict__ kernelD,  // (3,3)
                       const float* __restrict__ kernelP,  // (32,32)
                       const float* __restrict__ bias,     // (32)
                       const float* __restrict__ cvec,     // (1,8)
                       const float* __restrict__ mD_W1,    // (12,16)
                       const float* __restrict__ mD_b1,    // (16)
                       const float* __restrict__ mD_W2,    // (16,9)
                       const float* __restrict__ mD_b2,    // (9)
                       const float* __restrict__ mP_W1,    // (12,16)
                       const float* __restrict__ mP_b1,    // (16)
                       const float* __restrict__ mP_W2,    // (16,1024)
                       const float* __restrict__ mP_b2,    // (1024)
                       float* __restrict__ out)            // (8,32,128,128)
{
    __shared__ float hD_lds[TL * HID_];        // 1 KB
    __shared__ float hP_lds[TL * HID_];        // 1 KB
    __shared__ float kD_lds[TL * 9];           // 576 B
    __shared__ float y_lds[TL * B_ * CI];      // 16 KB  [l][b][i]
    __shared__ float kP_lds[TL * 16 * CO];     // 32 KB  [l][i_local][o]  (one i-half)

    const int tid  = threadIdx.x;
    const int wave = tid >> 5;
    const int lane = tid & 31;
    const int lm   = lane & 15;
    const int hiL  = lane >> 4;

    const int wg = blockIdx.x;            // 0..1023
    const int h  = wg >> 3;               // output row (reference quirk: xn uses h)
    const int w0 = (wg & 7) * TL;         // first output column of this tile

    // ---------------- Phase 1a: hidden layers of both MLPs ----------------
    {
        const int l = tid >> 4;           // location in tile (0..15)
        const int j = tid & 15;           // hidden unit
        const int w = w0 + l;
        const float xn = (float)h * (1.0f / 128.0f);
        const float yn = (float)w * (1.0f / 128.0f);
        float f[12];
        f[0] = xn; f[1] = 1.0f - xn; f[2] = yn; f[3] = 1.0f - yn;
        #pragma unroll
        for (int q = 0; q < 8; ++q) f[4 + q] = cvec[q];
        float aD = mD_b1[j], aP = mP_b1[j];
        #pragma unroll
        for (int q = 0; q < 12; ++q) {
            aD += f[q] * mD_W1[q * HID_ + j];
            aP += f[q] * mP_W1[q * HID_ + j];
        }
        hD_lds[l * HID_ + j] = fmaxf(aD, 0.0f);
        hP_lds[l * HID_ + j] = fmaxf(aP, 0.0f);
    }
    __syncthreads();

    // ---------------- Phase 1b: adaptive depthwise kernel kD ----------------
    if (tid < TL * 9) {
        const int l = tid / 9;
        const int k = tid % 9;
        float a = kernelD[k] + mD_b2[k];
        #pragma unroll
        for (int j = 0; j < HID_; ++j) a += hD_lds[l * HID_ + j] * mD_W2[j * 9 + k];
        kD_lds[l * 9 + k] = a;
    }
    __syncthreads();

    // ---------------- Phase 2: y[l][b][i] = depthwise(x, kD) ----------------
    {
        const int b = tid >> 5;           // 0..7
        const int i = tid & 31;           // 0..31
        const float* xp = x + (size_t)(b * CI + i) * (HH * WW);
        for (int l = 0; l < TL; ++l) {
            const int w = w0 + l;
            float acc = 0.0f;
            #pragma unroll
            for (int kh = 0; kh < 3; ++kh) {
                const int hh = h + kh - 1;
                if ((unsigned)hh < (unsigned)HH) {
                    #pragma unroll
                    for (int kw = 0; kw < 3; ++kw) {
                        const int ww = w + kw - 1;
                        if ((unsigned)ww < (unsigned)WW)
                            acc += xp[hh * WW + ww] * kD_lds[l * 9 + kh * 3 + kw];
                    }
                }
            }
            y_lds[(l * B_ + b) * CI + i] = acc;
        }
    }
    __syncthreads();

    // output accumulators: this wave owns 2 locations x 2 n-tiles (o 0-15, 16-31)
    v8f acc[2][2];
    #pragma unroll
    for (int a1 = 0; a1 < 2; ++a1)
        #pragma unroll
        for (int a2 = 0; a2 < 2; ++a2) acc[a1][a2] = (v8f){};

    // Process i (=C_IN) in two halves of 16 so kP tile fits in 32KB LDS.
    for (int ihalf = 0; ihalf < 2; ++ihalf) {
        // ---- Phase 3 (WMMA): kP_half[l][i_loc][o] = hP(16x16) @ W2cols + base ----
        // 512 local columns -> 32 n-tiles of 16; 8 waves x 4 tiles each.
        for (int t = 0; t < 4; ++t) {
            const int n    = (wave * 4 + t) * 16 + lm;  // local col 0..511
            const int gcol = ihalf * 512 + n;           // global col 0..1023
            v8f d = (v8f){};
            #pragma unroll
            for (int ks = 0; ks < 4; ++ks) {
                const int k0 = ks * 4 + 2 * hiL;        // K index for this lane-half
                v2f af, bf;
                af.x = hP_lds[lm * HID_ + k0];
                af.y = hP_lds[lm * HID_ + k0 + 1];
                bf.x = mP_W2[(size_t)k0 * 1024 + gcol];
                bf.y = mP_W2[(size_t)(k0 + 1) * 1024 + gcol];
                d = __builtin_amdgcn_wmma_f32_16x16x4_f32(
                        false, af, false, bf, (short)0, d, false, false);
            }
            const int i_loc = n >> 5;                   // local input-channel 0..15
            const int o     = n & 31;
            const int ig    = gcol >> 5;                // global input-channel
            const float kbase = kernelP[ig * CO + o] + mP_b2[gcol];
            #pragma unroll
            for (int r = 0; r < 8; ++r) {
                const int m = r + 8 * hiL;              // location row 0..15
                kP_lds[(m * 16 + i_loc) * CO + o] = d[r] + kbase;
            }
        }
        __syncthreads();

        // ---- Phase 4 (WMMA): out(16pad x 32) += Ypad(16x16) @ kP_l(16x32) ----
        #pragma unroll
        for (int li = 0; li < 2; ++li) {
            const int l = wave * 2 + li;
            #pragma unroll
            for (int nt = 0; nt < 2; ++nt) {
                v8f d = acc[li][nt];
                #pragma unroll
                for (int ks = 0; ks < 4; ++ks) {
                    const int k0 = ks * 4 + 2 * hiL;    // local i 0..15
                    v2f af, bf;
                    if (lm < 8) {                       // batch rows 8..15 are zero pad
                        af.x = y_lds[(l * B_ + lm) * CI + ihalf * 16 + k0];
                        af.y = y_lds[(l * B_ + lm) * CI + ihalf * 16 + k0 + 1];
                    } else {
                        af.x = 0.0f; af.y = 0.0f;
                    }
                    const int o = nt * 16 + lm;
                    bf.x = kP_lds[(l * 16 + k0) * CO + o];
                    bf.y = kP_lds[(l * 16 + k0 + 1) * CO + o];
                    d = __builtin_amdgcn_wmma_f32_16x16x4_f32(
                            false, af, false, bf, (short)0, d, false, false);
                }
                acc[li][nt] = d;
            }
        }
        __syncthreads();   // protect kP_lds reuse in next half
    }

    // ---------------- Store: D layout -> VGPR r holds M = r + 8*hiL ----------------
    #pragma unroll
    for (int li = 0; li < 2; ++li) {
        const int l = wave * 2 + li;
        const size_t lglob = (size_t)h * WW + (w0 + l);
        #pragma unroll
        for (int nt = 0; nt < 2; ++nt) {
            const int o = nt * 16 + lm;
            const float bo = bias[o];
            if (hiL == 0) {                 // only rows 0..7 are real batches
                #pragma unroll
                for (int r = 0; r < 8; ++r) {
                    out[(size_t)(r * CO + o) * (HH * WW) + lglob] = acc[li][nt][r] + bo;
                }
            }
        }
    }
}

extern "C" void kernel_launch(void* const* d_in, const int* in_sizes, int n_in,
                              void* d_out, int out_size, void* d_ws, size_t ws_size,
                              hipStream_t stream) {
    const float* x       = (const float*)d_in[0];
    const float* kernelD = (const float*)d_in[1];
    const float* kernelP = (const float*)d_in[2];
    const float* bias    = (const float*)d_in[3];
    const float* cvec    = (const float*)d_in[4];
    const float* mD_W1   = (const float*)d_in[5];
    const float* mD_b1   = (const float*)d_in[6];
    const float* mD_W2   = (const float*)d_in[7];
    const float* mD_b2   = (const float*)d_in[8];
    const float* mP_W1   = (const float*)d_in[9];
    const float* mP_b1   = (const float*)d_in[10];
    const float* mP_W2   = (const float*)d_in[11];
    const float* mP_b2   = (const float*)d_in[12];
    float* out = (float*)d_out;

    dim3 grid(1024);   // (128*128 locations) / 16 per workgroup
    dim3 block(256);   // 8 wave32 waves
    hipLaunchKernelGGL(scaconv_ds_kernel, grid, block, 0, stream,
                       x, kernelD, kernelP, bias, cvec,
                       mD_W1, mD_b1, mD_W2, mD_b2,
                       mP_W1, mP_b1, mP_W2, mP_b2, out);
}